// Renderer_83391085019277
// MI455X (gfx1250) — compile-verified
//
#include <hip/hip_runtime.h>
#include <stdint.h>

// ---------------- problem constants (match reference) ----------------
#define PP        48      // samples per ray
#define NKP       64      // keypoint bins per ray
#define OSTRIDE   69      // 1 mask + 1 depth + 3 channels + 64 kp
#define TILE      2048    // kp pairs per tile
#define KP_BLOCKS 1024
#define KP_THREADS 256
#define EPT       (TILE / KP_THREADS)   // 8 elements per thread per tile

// ---------------- order-preserving float<->uint keys ----------------
__device__ __forceinline__ unsigned f2key(float f) {
  unsigned u = __float_as_uint(f);
  return (u & 0x80000000u) ? ~u : (u | 0x80000000u);
}
__device__ __forceinline__ float key2f(unsigned k) {
  unsigned u = (k & 0x80000000u) ? (k & 0x7fffffffu) : ~k;
  return __uint_as_float(u);
}

__device__ __forceinline__ float wave_sum(float x) {
#pragma unroll
  for (int m = 16; m >= 1; m >>= 1) x += __shfl_xor(x, m, 32);
  return x;
}

// ---------------- kernel 1: zero output, init min/max keys ----------------
__global__ void init_out(float* __restrict__ out, int n, unsigned* __restrict__ keys) {
  int i = blockIdx.x * blockDim.x + threadIdx.x;
  if (i == 0) { keys[0] = 0xFFFFFFFFu; keys[1] = 0u; }
  for (; i < n; i += gridDim.x * blockDim.x) out[i] = 0.0f;
}

// ---------------- kernel 2: global min/max of depths ----------------
__global__ void depth_minmax(const float* __restrict__ d, int n, unsigned* __restrict__ keys) {
  float lmin = INFINITY, lmax = -INFINITY;
  for (int i = blockIdx.x * blockDim.x + threadIdx.x; i < n; i += gridDim.x * blockDim.x) {
    float v = d[i];
    lmin = fminf(lmin, v);
    lmax = fmaxf(lmax, v);
  }
#pragma unroll
  for (int m = 16; m >= 1; m >>= 1) {
    lmin = fminf(lmin, __shfl_xor(lmin, m, 32));
    lmax = fmaxf(lmax, __shfl_xor(lmax, m, 32));
  }
  if ((threadIdx.x & 31) == 0) {
    atomicMin(&keys[0], f2key(lmin));
    atomicMax(&keys[1], f2key(lmax));
  }
}

// ---------------- kernel 3: per-ray compositing (one wave32 per ray) ---------
// lanes 0..31 hold sample p=lane; lanes 0..15 also hold p=32+lane.
// transmittance = exclusive prefix product of (1 - alpha + 1e-10).
__global__ void composite_rays(const float* __restrict__ shp,
                               const float* __restrict__ dep,
                               const float* __restrict__ ch,
                               const int*   __restrict__ vm,
                               const unsigned* __restrict__ keys,
                               float* __restrict__ out,
                               float* __restrict__ wflat,
                               int nRays) {
  int ray  = (int)((blockIdx.x * blockDim.x + threadIdx.x) >> 5);
  int lane = threadIdx.x & 31;
  if (ray >= nRays) return;
  long base = (long)ray * PP;
  bool hi = lane < (PP - 32);            // lane < 16

  float x0 = shp[base + lane];
  float d0 = dep[base + lane];
  int   v0 = vm [base + lane];
  float x1 = 0.f, d1 = 0.f; int v1 = 0;
  if (hi) { x1 = shp[base + 32 + lane]; d1 = dep[base + 32 + lane]; v1 = vm[base + 32 + lane]; }

  float a0 = 1.0f / (1.0f + __expf(-x0));
  float a1 = hi ? 1.0f / (1.0f + __expf(-x1)) : 0.0f;
  float s0 = 1.0f - a0 + 1e-10f;
  float s1 = hi ? (1.0f - a1 + 1e-10f) : 1.0f;

  // inclusive prefix product over 32 lanes
  float inc0 = s0;
#pragma unroll
  for (int dlt = 1; dlt < 32; dlt <<= 1) {
    float o = __shfl_up(inc0, dlt, 32);
    if (lane >= dlt) inc0 *= o;
  }
  float tr0  = (lane == 0) ? 1.0f : __shfl_up(inc0, 1, 32);   // exclusive
  float tot0 = __shfl(inc0, 31, 32);                          // product of first 32
  // scan the upper 16 samples (lanes 0..15 only pull from lower lanes)
  float inc1 = s1;
#pragma unroll
  for (int dlt = 1; dlt < 16; dlt <<= 1) {
    float o = __shfl_up(inc1, dlt, 32);
    if (lane >= dlt) inc1 *= o;
  }
  float tr1 = ((lane == 0) ? 1.0f : __shfl_up(inc1, 1, 32)) * tot0;

  float w0 = a0 * tr0;
  float w1 = a1 * tr1;                    // a1==0 for lane>=16
  float wv0 = v0 ? w0 : 0.0f;
  float wv1 = (hi && v1) ? w1 : 0.0f;

  // spill masked weights for the keypoint scatter pass (coalesced)
  wflat[base + lane] = wv0;
  if (hi) wflat[base + 32 + lane] = wv1;

  long cb0 = (base + lane) * 3;
  float c0 = wv0 * ch[cb0 + 0];
  float c1 = wv0 * ch[cb0 + 1];
  float c2 = wv0 * ch[cb0 + 2];
  if (hi) {
    long cb1 = (base + 32 + lane) * 3;
    c0 += wv1 * ch[cb1 + 0];
    c1 += wv1 * ch[cb1 + 1];
    c2 += wv1 * ch[cb1 + 2];
  }

  float wt = wave_sum(w0 + w1);                       // unmasked (reference)
  float wd = wave_sum(w0 * d0 + w1 * d1);
  c0 = wave_sum(c0); c1 = wave_sum(c1); c2 = wave_sum(c2);

  if (lane == 0) {
    float gmin = key2f(keys[0]);
    float gmax = key2f(keys[1]);
    float depth = wd / wt;
    if (depth != depth) depth = INFINITY;             // NaN -> inf
    depth = fminf(fmaxf(depth, gmin), gmax);
    float* o = out + (long)ray * OSTRIDE;
    o[0] = fminf(fmaxf(wt, 0.0f), 1.0f);
    o[1] = depth;
    o[2] = fminf(fmaxf(c0, 0.0f), 1.0f);
    o[3] = fminf(fmaxf(c1, 0.0f), 1.0f);
    o[4] = fminf(fmaxf(c2, 0.0f), 1.0f);
  }
}

// ---------------- kernel 4: keypoint scatter with async-to-LDS staging -------
// Streams kp_w / kp_shading_idx / kp_kp_idx through LDS with
// GLOBAL_LOAD_ASYNC_TO_LDS_B32 (ASYNCcnt), double-buffered. Each thread only
// consumes its own LDS slots -> no barriers; per-wave s_wait_asynccnt suffices.
__device__ __forceinline__ void async_b32(unsigned lds_addr, const void* gaddr) {
  asm volatile("global_load_async_to_lds_b32 %0, %1, off"
               :: "v"(lds_addr), "v"((unsigned long long)(uintptr_t)gaddr)
               : "memory");
}

__global__ void kp_scatter(const float* __restrict__ kpw,
                           const int*   __restrict__ sidx,
                           const int*   __restrict__ kidx,
                           const float* __restrict__ wflat,
                           float* __restrict__ out,
                           int nPairs) {
  __shared__ float sW[2][TILE];
  __shared__ int   sS[2][TILE];
  __shared__ int   sK[2][TILE];
  const int tid = threadIdx.x;
  const int tilesTotal    = nPairs / TILE;
  const int tilesPerBlock = tilesTotal / gridDim.x;
  const long tile0 = (long)blockIdx.x * tilesPerBlock;

  auto issue = [&](int t, int b) {
    long gb = (tile0 + t) * TILE;
#pragma unroll
    for (int k = 0; k < EPT; ++k) {
      int j = k * KP_THREADS + tid;
      async_b32((unsigned)(uintptr_t)&sW[b][j], kpw  + gb + j);
      async_b32((unsigned)(uintptr_t)&sS[b][j], sidx + gb + j);
      async_b32((unsigned)(uintptr_t)&sK[b][j], kidx + gb + j);
    }
  };

  if (tilesPerBlock > 0) issue(0, 0);
  for (int t = 0; t < tilesPerBlock; ++t) {
    int b = t & 1;
    bool more = (t + 1) < tilesPerBlock;
    if (more) {
      issue(t + 1, 1 - b);
      asm volatile("s_wait_asynccnt 24" ::: "memory");  // tile t complete, t+1 in flight
    } else {
      asm volatile("s_wait_asynccnt 0" ::: "memory");
    }
#pragma unroll
    for (int k = 0; k < EPT; ++k) {
      int j = k * KP_THREADS + tid;
      float w  = sW[b][j];
      int   si = sS[b][j];
      int   ki = sK[b][j];
      float val = wflat[si] * w;                 // gather: L2-resident (12.6 MB)
      if (val != 0.0f) {
        unsigned ray = (unsigned)si / (unsigned)PP;
        atomicAdd(out + (unsigned long long)ray * OSTRIDE + 5u + (unsigned)ki, val);
      }
    }
  }
  // tail (nPairs not tile-divisible): grid-stride direct path
  long done = (long)tilesTotal * TILE;
  for (long i = done + blockIdx.x * blockDim.x + tid; i < nPairs;
       i += (long)gridDim.x * blockDim.x) {
    float val = wflat[sidx[i]] * kpw[i];
    if (val != 0.0f) {
      unsigned ray = (unsigned)sidx[i] / (unsigned)PP;
      atomicAdd(out + (unsigned long long)ray * OSTRIDE + 5u + (unsigned)kidx[i], val);
    }
  }
}

// ---------------- kernel 5: clip keypoint bins ----------------
__global__ void kp_clip(float* __restrict__ out, int nRays) {
  int n = nRays * NKP;
  for (int i = blockIdx.x * blockDim.x + threadIdx.x; i < n; i += gridDim.x * blockDim.x) {
    int ray = i >> 6, k = i & 63;
    float* p = out + (long)ray * OSTRIDE + 5 + k;
    *p = fminf(fmaxf(*p, 0.0f), 1.0f);
  }
}

// ---------------- launcher ----------------
extern "C" void kernel_launch(void* const* d_in, const int* in_sizes, int n_in,
                              void* d_out, int out_size, void* d_ws, size_t ws_size,
                              hipStream_t stream) {
  const float* shp  = (const float*)d_in[0];   // shape
  const float* dep  = (const float*)d_in[1];   // depths
  const float* ch   = (const float*)d_in[2];   // channels [pts,3]
  const float* kpw  = (const float*)d_in[3];   // kp_w
  const int*   sidx = (const int*)  d_in[4];   // kp_shading_idx
  const int*   kidx = (const int*)  d_in[5];   // kp_kp_idx
  const int*   vm   = (const int*)  d_in[6];   // valid_mask

  const int nPoints = in_sizes[0];
  const int nPairs  = in_sizes[3];
  const int nRays   = nPoints / PP;

  float*    out   = (float*)d_out;
  unsigned* keys  = (unsigned*)d_ws;                    // 2 x u32 min/max keys
  float*    wflat = (float*)((char*)d_ws + 256);        // nPoints masked weights

  init_out    <<<2048, 256, 0, stream>>>(out, out_size, keys);
  depth_minmax<<<1024, 256, 0, stream>>>(dep, nPoints, keys);
  composite_rays<<<(nRays + 7) / 8, 256, 0, stream>>>(shp, dep, ch, vm, keys,
                                                      out, wflat, nRays);
  kp_scatter  <<<KP_BLOCKS, KP_THREADS, 0, stream>>>(kpw, sidx, kidx, wflat, out, nPairs);
  kp_clip     <<<2048, 256, 0, stream>>>(out, nRays);
}